// Gcn2_29858612642432
// MI455X (gfx1250) — compile-verified
//
#include <hip/hip_runtime.h>
#include <math.h>

// ---------------------------------------------------------------------------
// GCN forward for MI455X (gfx1250, wave32).
// Dense GEMMs use V_WMMA_F32_16X16X4_F32 (fp32 A/B, fp32 acc).
// Edge aggregation = float4 gather + global_atomic_add_f32 scatter (L2-resident).
// ---------------------------------------------------------------------------

typedef __attribute__((ext_vector_type(2))) float v2f;
typedef __attribute__((ext_vector_type(8))) float v8f;

#define DFEAT 64      // D_IN == D_H == 64
#define WP_STRIDE 34  // padded float2 stride per column: banks (4c+2k)%64 -> conflict-free

// ---------------------------------------------------------------------------
// Degree kernels
// ---------------------------------------------------------------------------
__global__ void deg_count_kernel(float* __restrict__ deg,
                                 const int* __restrict__ dst, int e) {
  int t = blockIdx.x * blockDim.x + threadIdx.x;
  if (t < e) atomicAdd(&deg[dst[t]], 1.0f);
}

__global__ void deg_finalize_kernel(float* __restrict__ deg, int n) {
  int t = blockIdx.x * blockDim.x + threadIdx.x;
  if (t < n) deg[t] = rsqrtf(deg[t] + 1.0f);  // deg over A+I; becomes dinv
}

// ---------------------------------------------------------------------------
// WMMA GEMM:  Y[n x DCOL] = X[n x 64] @ W[64 x DCOL]  (+ bias if BIAS)
//
// One wave per 16-row tile, 8 waves/block. K-loop steps of 4 using
// V_WMMA_F32_16X16X4_F32.
//
// f32 WMMA VGPR layouts (ISA 7.12.2):
//   A 16x4 : lane l: row = l&15, K pair selected by l>>4 ->
//            v2f {A[r][ka], A[r][ka+1]}, ka = k0 + 2*(l>>4)   (one b64 load)
//   B 4x16 : mirrored: v2f {W[ka][c], W[ka+1][c]}, c = l&15
//            -> staged in LDS transposed + pair-packed so this is ONE aligned
//               ds_load_b64 into an even register pair (no repack movs):
//               wp[c*WP_STRIDE + k/2] = {W[k][c], W[k+1][c]}
//   C/D    : 8 VGPRs; M = v + 8*(l>>4), N = l&15
// ---------------------------------------------------------------------------
template <int DCOL, bool BIAS>
__global__ void gemm_wmma_kernel(const float* __restrict__ X,
                                 const float* __restrict__ W,
                                 const float* __restrict__ bias,
                                 float* __restrict__ Y, int n) {
  __shared__ v2f wp[DCOL * WP_STRIDE];

  // Stage W[64 x DCOL] transposed + K-pair-packed.
  for (int i = threadIdx.x; i < DCOL * (DFEAT / 2); i += 256) {
    const int col = i / (DFEAT / 2);
    const int kp  = i % (DFEAT / 2);
    v2f w;
    w.x = W[(2 * kp) * DCOL + col];
    w.y = W[(2 * kp + 1) * DCOL + col];
    wp[col * WP_STRIDE + kp] = w;
  }
  __syncthreads();

  const int wave  = threadIdx.x >> 5;
  const int lane  = threadIdx.x & 31;
  const int tile  = blockIdx.x * 8 + wave;
  const int tiles = (n + 15) >> 4;
  if (tile >= tiles) return;  // uniform per wave: EXEC stays all-ones

  const int half = lane >> 4;  // 0 or 1 (K-pair / M-half select)
  const int l16  = lane & 15;

  const int row  = tile * 16 + l16;
  const int rowc = row < n ? row : n - 1;  // clamp load, keep EXEC full
  const float* __restrict__ xrow = X + (size_t)rowc * DFEAT;

  constexpr int NT = DCOL / 16;
  v8f acc[NT] = {};

#pragma unroll
  for (int k0 = 0; k0 < DFEAT; k0 += 4) {
    const int ka   = k0 + half * 2;
    const int kidx = (k0 >> 1) + half;
    const v2f a = *(const v2f*)(xrow + ka);  // 8B-aligned (ka even)
#pragma unroll
    for (int t = 0; t < NT; ++t) {
      const int col = t * 16 + l16;
      const v2f b = wp[col * WP_STRIDE + kidx];  // single ds_load_b64
      acc[t] = __builtin_amdgcn_wmma_f32_16x16x4_f32(
          false, a, false, b, (short)0, acc[t], false, false);
    }
  }

  // Store D (divergent guard only after all WMMAs have issued).
#pragma unroll
  for (int t = 0; t < NT; ++t) {
    const float bb = BIAS ? bias[t * 16 + l16] : 0.0f;
#pragma unroll
    for (int v = 0; v < 8; ++v) {
      const int orow = tile * 16 + half * 8 + v;
      if (orow < n) Y[(size_t)orow * DCOL + t * 16 + l16] = acc[t][v] + bb;
    }
  }
}

// ---------------------------------------------------------------------------
// Edge aggregation: agg[dst] += h[src] * (dinv[src]*dinv[dst])
// 16 threads per edge, float4 per thread -> 4 f32 atomics (L2-resident RMW).
// ---------------------------------------------------------------------------
__global__ void aggregate_kernel(float* __restrict__ agg,
                                 const float* __restrict__ h,
                                 const int* __restrict__ src,
                                 const int* __restrict__ dst,
                                 const float* __restrict__ dinv, int e) {
  const int tid  = blockIdx.x * blockDim.x + threadIdx.x;
  const int edge = tid >> 4;
  if (edge >= e) return;
  const int c0 = (tid & 15) * 4;
  const int s  = src[edge];
  const int d  = dst[edge];
  const float norm = dinv[s] * dinv[d];
  const float4 hv = *(const float4*)(h + (size_t)s * DFEAT + c0);
  float* base = agg + (size_t)d * DFEAT + c0;
  atomicAdd(base + 0, hv.x * norm);
  atomicAdd(base + 1, hv.y * norm);
  atomicAdd(base + 2, hv.z * norm);
  atomicAdd(base + 3, hv.w * norm);
}

// ---------------------------------------------------------------------------
// Self-loop + bias + tanh:  cur = tanh(agg + h * dinv^2 + b)
// ---------------------------------------------------------------------------
__global__ void finalize_kernel(float* __restrict__ out,
                                const float* __restrict__ agg,
                                const float* __restrict__ h,
                                const float* __restrict__ dinv,
                                const float* __restrict__ bias, int n) {
  const int t = blockIdx.x * blockDim.x + threadIdx.x;
  if (t >= n * DFEAT) return;
  const int i = t >> 6;
  const int c = t & (DFEAT - 1);
  const float di = dinv[i];
  out[t] = tanhf(agg[t] + h[t] * di * di + bias[c]);
}

// ---------------------------------------------------------------------------
// Launch
// ---------------------------------------------------------------------------
extern "C" void kernel_launch(void* const* d_in, const int* in_sizes, int n_in,
                              void* d_out, int out_size, void* d_ws,
                              size_t ws_size, hipStream_t stream) {
  const float* x  = (const float*)d_in[0];
  const int*   ei = (const int*)d_in[1];  // [2, E]
  const float* Ws[4] = {(const float*)d_in[2], (const float*)d_in[4],
                        (const float*)d_in[6], (const float*)d_in[8]};
  const float* bs[4] = {(const float*)d_in[3], (const float*)d_in[5],
                        (const float*)d_in[7], (const float*)d_in[9]};
  const float* W_out = (const float*)d_in[10];
  const float* b_out = (const float*)d_in[11];

  const int n = in_sizes[0] / DFEAT;  // 100000
  const int e = in_sizes[1] / 2;      // 1000000
  const int* src = ei;
  const int* dst = ei + e;

  // Workspace layout (fp32): dinv[n] | cur[n*64] | h[n*64] | agg[n*64]
  float* dinv = (float*)d_ws;
  float* cur  = dinv + n;
  float* h    = cur + (size_t)n * DFEAT;
  float* agg  = h + (size_t)n * DFEAT;

  const int tiles      = (n + 15) / 16;
  const int gemmBlocks = (tiles + 7) / 8;  // 8 waves (16-row tiles) per block

  // 1) degrees -> dinv = rsqrt(deg + 1)
  hipMemsetAsync(dinv, 0, (size_t)n * sizeof(float), stream);
  deg_count_kernel<<<(e + 255) / 256, 256, 0, stream>>>(dinv, dst, e);
  deg_finalize_kernel<<<(n + 255) / 256, 256, 0, stream>>>(dinv, n);

  // 2) four GCN layers
  for (int l = 0; l < 4; ++l) {
    const float* in = (l == 0) ? x : cur;
    gemm_wmma_kernel<DFEAT, false>
        <<<gemmBlocks, 256, 0, stream>>>(in, Ws[l], nullptr, h, n);
    hipMemsetAsync(agg, 0, (size_t)n * DFEAT * sizeof(float), stream);
    aggregate_kernel<<<(e * 16 + 255) / 256, 256, 0, stream>>>(agg, h, src,
                                                               dst, dinv, e);
    finalize_kernel<<<((size_t)n * DFEAT + 255) / 256, 256, 0, stream>>>(
        cur, agg, h, dinv, bs[l], n);
  }

  // 3) output projection with fused bias
  gemm_wmma_kernel<16, true>
      <<<gemmBlocks, 256, 0, stream>>>(cur, W_out, b_out, (float*)d_out, n);
}